// MultiNet_56521769615917
// MI455X (gfx1250) — compile-verified
//
#include <hip/hip_runtime.h>
#include <math.h>

typedef __attribute__((ext_vector_type(2))) float v2f;
typedef __attribute__((ext_vector_type(8))) float v8f;

// ---------------- degree / elementwise kernels ----------------

__global__ void k_fill1(float* __restrict__ p, int n) {
  int i = blockIdx.x * 256 + threadIdx.x;
  if (i < n) p[i] = 1.0f;  // self-loop contribution to degree
}

__global__ void k_deg_edges(float* __restrict__ deg, const int* __restrict__ dst, int E) {
  int i = blockIdx.x * 256 + threadIdx.x;
  if (i < E) atomicAdd(&deg[dst[i]], 1.0f);
}

__global__ void k_rsqrt(float* __restrict__ p, int n) {
  int i = blockIdx.x * 256 + threadIdx.x;
  if (i < n) p[i] = rsqrtf(p[i]);  // deg >= 1 always (self loop)
}

// layer-1 value: t[i] = x[i]*dinv[i]; accumulator starts at self-loop value
__global__ void k_l1_init(const float* __restrict__ x, const float* __restrict__ dinv,
                          float* __restrict__ tbuf, float* __restrict__ sbuf, int n) {
  int i = blockIdx.x * 256 + threadIdx.x;
  if (i < n) { float v = x[i] * dinv[i]; tbuf[i] = v; sbuf[i] = v; }
}

// generic scalar scatter-add: sbuf[dst] += tbuf[src]
__global__ void k_scatter_scalar(const float* __restrict__ tbuf, float* __restrict__ sbuf,
                                 const int* __restrict__ src, const int* __restrict__ dst, int E) {
  int i = blockIdx.x * 256 + threadIdx.x;
  if (i < E) atomicAdd(&sbuf[dst[i]], tbuf[src[i]]);
}

// h1[i][k] = relu(sbuf[i]*dinv[i]*W_in[k] + b_in[k])   (W_in factored out of segment-sum)
__global__ void k_l1_fin(const float* __restrict__ sbuf, const float* __restrict__ dinv,
                         const float* __restrict__ Win, const float* __restrict__ bin,
                         float* __restrict__ HA, int n) {
  int idx = blockIdx.x * 256 + threadIdx.x;
  if (idx >= n * 8) return;
  int node = idx >> 3, g = idx & 7;
  float s = sbuf[node] * dinv[node];
  float4 w = ((const float4*)Win)[g];
  float4 b = ((const float4*)bin)[g];
  float4 r;
  r.x = fmaxf(fmaf(s, w.x, b.x), 0.0f);
  r.y = fmaxf(fmaf(s, w.y, b.y), 0.0f);
  r.z = fmaxf(fmaf(s, w.z, b.z), 0.0f);
  r.w = fmaxf(fmaf(s, w.w, b.w), 0.0f);
  ((float4*)HA)[(size_t)node * 8 + g] = r;
}

// ---------------- WMMA fp32 GEMM: T = (H @ W) * dinv[row]; AGG init = T ----------------
// One wave handles 16 nodes. Full fp32 via V_WMMA_F32_16X16X4_F32, K=32 in 8 steps,
// two 16-wide column tiles. dinv source-scaling + self-loop agg-init fused in epilogue.
__global__ void k_gemm32(const float* __restrict__ H, const float* __restrict__ W,
                         const float* __restrict__ dinv, float* __restrict__ T,
                         float* __restrict__ AGG, int n) {
  int gw   = (blockIdx.x * blockDim.x + threadIdx.x) >> 5;  // global wave id = tile id
  int lane = threadIdx.x & 31;
  int ntiles = (n + 15) >> 4;
  if (gw >= ntiles) return;            // wave-uniform exit: EXEC stays all-1s for WMMA
  int r  = lane & 15;
  int hi = lane >> 4;                  // lane-half selects K offset {0,1} vs {2,3}
  int base = gw * 16;
  int row = base + r;
  if (row >= n) row = n - 1;           // clamped loads; stores for OOB rows are skipped

  // A fragments: a[s] = H[row][4s+2hi .. +1]
  v2f a[8];
  const float2* h2 = (const float2*)(H + (size_t)row * 32);
  #pragma unroll
  for (int s = 0; s < 8; ++s) {
    float2 t = h2[2 * s + hi];
    a[s][0] = t.x; a[s][1] = t.y;
  }
  // B fragments: b[c][s] = W[4s+2hi .. +1][16c + r]
  v2f b0[8], b1[8];
  #pragma unroll
  for (int s = 0; s < 8; ++s) {
    int k = 4 * s + 2 * hi;
    b0[s][0] = W[(size_t)k * 32 + r];        b0[s][1] = W[(size_t)(k + 1) * 32 + r];
    b1[s][0] = W[(size_t)k * 32 + 16 + r];   b1[s][1] = W[(size_t)(k + 1) * 32 + 16 + r];
  }

  v8f acc0 = {}, acc1 = {};
  #pragma unroll
  for (int s = 0; s < 8; ++s) {
    acc0 = __builtin_amdgcn_wmma_f32_16x16x4_f32(false, a[s], false, b0[s], (short)0, acc0, false, false);
    acc1 = __builtin_amdgcn_wmma_f32_16x16x4_f32(false, a[s], false, b1[s], (short)0, acc1, false, false);
  }

  // D layout: VGPR v -> row (base + v + 8*hi), col r (tile 0) / 16+r (tile 1)
  if (base + 16 <= n) {
    // full tile (always the case when n % 16 == 0): branch-free epilogue,
    // dinv for this lane's 8 rows fetched with two float4 loads
    const float4* dv4 = (const float4*)(dinv + base + 8 * hi);
    float4 d0 = dv4[0], d1 = dv4[1];
    float dm[8] = {d0.x, d0.y, d0.z, d0.w, d1.x, d1.y, d1.z, d1.w};
    #pragma unroll
    for (int v = 0; v < 8; ++v) {
      int m = base + v + 8 * hi;
      float x0 = acc0[v] * dm[v], x1 = acc1[v] * dm[v];
      size_t o0 = (size_t)m * 32 + r;
      T[o0] = x0;  T[o0 + 16] = x1;    // gather source for edge pass
      AGG[o0] = x0; AGG[o0 + 16] = x1; // self-loop term = accumulator init
    }
  } else {
    #pragma unroll
    for (int v = 0; v < 8; ++v) {
      int m = base + v + 8 * hi;
      if (m < n) {
        float dmv = dinv[m];
        float x0 = acc0[v] * dmv, x1 = acc1[v] * dmv;
        size_t o0 = (size_t)m * 32 + r;
        T[o0] = x0;  T[o0 + 16] = x1;
        AGG[o0] = x0; AGG[o0 + 16] = x1;
      }
    }
  }
}

// 32-wide edge scatter: 8 lanes per edge, float4 gather + 4 f32 atomics each
__global__ void k_scatter32(const float* __restrict__ T, float* __restrict__ AGG,
                            const int* __restrict__ src, const int* __restrict__ dst,
                            long long total8) {
  long long t = (long long)blockIdx.x * 256 + threadIdx.x;
  if (t >= total8) return;
  int e = (int)(t >> 3);
  int g = (int)(t & 7);
  int s = src[e], d = dst[e];
  float4 v = *(const float4*)(T + (size_t)s * 32 + g * 4);
  float* o = AGG + (size_t)d * 32 + g * 4;
  atomicAdd(o + 0, v.x);
  atomicAdd(o + 1, v.y);
  atomicAdd(o + 2, v.z);
  atomicAdd(o + 3, v.w);
}

// H[i][k] = relu(dinv[i]*AGG[i][k] + b[k])   (in place on HA)
__global__ void k_finalize(float* __restrict__ HA, const float* __restrict__ dinv,
                           const float* __restrict__ bmid, int n) {
  int idx = blockIdx.x * 256 + threadIdx.x;
  if (idx >= n * 8) return;
  int node = idx >> 3, g = idx & 7;
  float d = dinv[node];
  float4 a = ((const float4*)HA)[(size_t)node * 8 + g];
  float4 b = ((const float4*)bmid)[g];
  float4 r;
  r.x = fmaxf(fmaf(d, a.x, b.x), 0.0f);
  r.y = fmaxf(fmaf(d, a.y, b.y), 0.0f);
  r.z = fmaxf(fmaf(d, a.z, b.z), 0.0f);
  r.w = fmaxf(fmaf(d, a.w, b.w), 0.0f);
  ((float4*)HA)[(size_t)node * 8 + g] = r;
}

// output layer 32->1: tbuf[i] = dinv[i] * dot(H[i], W_out); sbuf init = self-loop term
__global__ void k_dot(const float* __restrict__ H, const float* __restrict__ Wout,
                      const float* __restrict__ dinv,
                      float* __restrict__ tbuf, float* __restrict__ sbuf, int n) {
  int i = blockIdx.x * 256 + threadIdx.x;
  if (i >= n) return;
  const float4* h4 = (const float4*)(H + (size_t)i * 32);
  const float4* w4 = (const float4*)Wout;
  float acc = 0.0f;
  #pragma unroll
  for (int g = 0; g < 8; ++g) {
    float4 h = h4[g], w = w4[g];
    acc = fmaf(h.x, w.x, acc); acc = fmaf(h.y, w.y, acc);
    acc = fmaf(h.z, w.z, acc); acc = fmaf(h.w, w.w, acc);
  }
  float v = acc * dinv[i];
  tbuf[i] = v; sbuf[i] = v;
}

__global__ void k_sigmoid(const float* __restrict__ sbuf, const float* __restrict__ dinv,
                          const float* __restrict__ bout, float* __restrict__ out, int n) {
  int i = blockIdx.x * 256 + threadIdx.x;
  if (i >= n) return;
  float z = fmaf(dinv[i], sbuf[i], bout[0]);
  out[i] = 1.0f / (1.0f + __expf(-z));
}

// ---------------- launch ----------------

static inline int cdiv(long long a, long long b) { return (int)((a + b - 1) / b); }

extern "C" void kernel_launch(void* const* d_in, const int* in_sizes, int n_in,
                              void* d_out, int out_size, void* d_ws, size_t ws_size,
                              hipStream_t stream) {
  const float* x    = (const float*)d_in[0];
  const int*   ei   = (const int*)  d_in[1];
  const float* Win  = (const float*)d_in[2];
  const float* bin  = (const float*)d_in[3];
  const float* Wmid = (const float*)d_in[4];
  const float* bmid = (const float*)d_in[5];
  const float* Wout = (const float*)d_in[6];
  const float* bout = (const float*)d_in[7];
  float* out = (float*)d_out;

  int N = in_sizes[0];
  int E = in_sizes[1] / 2;
  const int* src = ei;
  const int* dst = ei + E;

  float* ws   = (float*)d_ws;
  float* dinv = ws;                 // N
  float* sbuf = ws + (size_t)N;     // N  (scalar accumulator)
  float* tbuf = ws + 2 * (size_t)N; // N  (scalar values)
  size_t off  = (3 * (size_t)N + 3) & ~(size_t)3;  // 16B align
  float* HA   = ws + off;                      // N x 32 features (doubles as agg target)
  float* HB   = HA + 32 * (size_t)N;           // N x 32 transformed (gather source)

  // degree with self-loops -> dinv = rsqrt(deg)
  k_fill1    <<<cdiv(N, 256), 256, 0, stream>>>(dinv, N);
  k_deg_edges<<<cdiv(E, 256), 256, 0, stream>>>(dinv, dst, E);
  k_rsqrt    <<<cdiv(N, 256), 256, 0, stream>>>(dinv, N);

  // layer 1 (1 -> 32): scalar aggregation, W_in factored out
  k_l1_init       <<<cdiv(N, 256), 256, 0, stream>>>(x, dinv, tbuf, sbuf, N);
  k_scatter_scalar<<<cdiv(E, 256), 256, 0, stream>>>(tbuf, sbuf, src, dst, E);
  k_l1_fin        <<<cdiv((long long)N * 8, 256), 256, 0, stream>>>(sbuf, dinv, Win, bin, HA, N);

  // two middle layers (32 -> 32), shared W_mid
  int ntiles = (N + 15) / 16;
  long long total8 = (long long)E * 8;
  for (int l = 0; l < 2; ++l) {
    k_gemm32  <<<cdiv((long long)ntiles * 32, 256), 256, 0, stream>>>(HA, Wmid, dinv, HB, HA, N);
    k_scatter32<<<cdiv(total8, 256), 256, 0, stream>>>(HB, HA, src, dst, total8);
    k_finalize<<<cdiv((long long)N * 8, 256), 256, 0, stream>>>(HA, dinv, bmid, N);
  }

  // output layer (32 -> 1): scalar aggregation again, then sigmoid
  k_dot           <<<cdiv(N, 256), 256, 0, stream>>>(HA, Wout, dinv, tbuf, sbuf, N);
  k_scatter_scalar<<<cdiv(E, 256), 256, 0, stream>>>(tbuf, sbuf, src, dst, E);
  k_sigmoid       <<<cdiv(N, 256), 256, 0, stream>>>(sbuf, dinv, bout, out, N);
}